// WormholeGather_84430467105120
// MI455X (gfx1250) — compile-verified
//
#include <hip/hip_runtime.h>
#include <stdint.h>

// ---------------------------------------------------------------------------
// WormholeGather on MI455X (gfx1250)
//   out[b,p,:] = sum_k weights[b,p,k] * x[b, routes[b,p,k], :]
// Strategy: one wave32 per (b,p). The K=16 routed rows (16KB) are gathered by
// the Tensor Data Mover in ONE tensor_load_to_lds (gather_mode=1, 16-bit row
// indices), then the wave does the weighted reduction out of LDS with
// wave-uniform (SGPR) weights, and writes one coalesced 1KB output row.
// Memory-bound workload (~0.5 FLOP/byte): WMMA is inapplicable (B operand of
// the would-be matmul depends on both p and k), TDM gather mode is the
// purpose-built CDNA5 path for exactly this indexed-row gather.
// ---------------------------------------------------------------------------

typedef unsigned int u32x4 __attribute__((ext_vector_type(4)));
typedef int          i32x8 __attribute__((ext_vector_type(8)));
typedef int          i32x4 __attribute__((ext_vector_type(4)));

#define B_DIM 4
#define P_DIM 8192
#define K_DIM 16
#define D_DIM 256
#define WAVES_PER_BLOCK 4

__global__ __launch_bounds__(32 * WAVES_PER_BLOCK)
void wormhole_gather_tdm_kernel(const float* __restrict__ x,
                                const int*   __restrict__ routes,
                                const float* __restrict__ weights,
                                float*       __restrict__ out) {
  // 16KB gather tile per wave: [K rows][D floats], 64KB per workgroup.
  // 320KB LDS/WGP -> 5 workgroups = 20 waves per WGP to hide DMA latency.
  __shared__ float tile[WAVES_PER_BLOCK][K_DIM][D_DIM];

  // Wave id as a compiler-provable scalar (SGPR) so the whole descriptor and
  // the routes/weights pointers are uniform -> s_load path.
  const int tid0 = __builtin_amdgcn_readfirstlane((int)threadIdx.x);
  const int w    = tid0 >> 5;                       // wave in block (uniform)
  const int lane = (int)(threadIdx.x & 31u);
  const int g    = (int)blockIdx.x * WAVES_PER_BLOCK + w;  // flat (b,p), uniform
  const int b    = g >> 13;                         // g / P_DIM

  const int*   rt = routes  + (size_t)g * K_DIM;    // uniform pointers
  const float* wt = weights + (size_t)g * K_DIM;

#if __has_builtin(__builtin_amdgcn_tensor_load_to_lds)
  // ---- Tensor DMA descriptor (D#), gather mode, 16-bit indices ------------
  const uint64_t xaddr  = (uint64_t)(uintptr_t)(x + (size_t)b * P_DIM * D_DIM);
  // Low 32 bits of a generic pointer to LDS == workgroup-relative LDS offset
  // (ISA 10.2).
  const uint32_t ldsoff = (uint32_t)(uintptr_t)(void*)&tile[w][0][0];

  u32x4 d0;
  d0[0] = 0x80000001u;                       // gather_mode=1, idx16, count=1
  d0[1] = ldsoff;                            // lds_addr (bytes)
  d0[2] = (uint32_t)xaddr;                   // global_addr[31:0]
  d0[3] = ((uint32_t)(xaddr >> 32) & 0x01FFFFFFu) | (2u << 30); // addr hi, type=2

  i32x8 d1;
  d1[0] = 0x00020000;                        // data_size=2 (4B), wg_mask=0
  d1[1] = (int)(((uint32_t)D_DIM & 0xFFFFu) << 16); // tensor_dim0[15:0] @ [63:48]
  d1[2] = (int)(((uint32_t)P_DIM & 0xFFFFu) << 16); // tensor_dim1[15:0] @ [95:80]
  d1[3] = (int)(((uint32_t)D_DIM & 0xFFFFu) << 16); // tile_dim0 @ [127:112]
  d1[4] = K_DIM;                             // tile_dim1 = #valid indices
  d1[5] = D_DIM;                             // tensor_dim0_stride[31:0]
  d1[6] = 0;                                 // stride hi / dim1_stride (ignored)
  d1[7] = 0;

  // Groups 2/3: the 16 row indices, two 16-bit indices per dword.
  i32x4 d2, d3;
  d2[0] = (rt[0]  & 0xFFFF) | (rt[1]  << 16);
  d2[1] = (rt[2]  & 0xFFFF) | (rt[3]  << 16);
  d2[2] = (rt[4]  & 0xFFFF) | (rt[5]  << 16);
  d2[3] = (rt[6]  & 0xFFFF) | (rt[7]  << 16);
  d3[0] = (rt[8]  & 0xFFFF) | (rt[9]  << 16);
  d3[1] = (rt[10] & 0xFFFF) | (rt[11] << 16);
  d3[2] = (rt[12] & 0xFFFF) | (rt[13] << 16);
  d3[3] = (rt[14] & 0xFFFF) | (rt[15] << 16);

  // 5th SGPR-group slot (VIMAGE VADDR4 is unused per ISA -> zero-filled).
  i32x8 d4;
  d4[0] = 0; d4[1] = 0; d4[2] = 0; d4[3] = 0;
  d4[4] = 0; d4[5] = 0; d4[6] = 0; d4[7] = 0;

  __builtin_amdgcn_tensor_load_to_lds(d0, d1, d2, d3, d4, 0);
  __builtin_amdgcn_s_wait_tensorcnt(0);      // 16KB tile now resident in LDS
#else
  // Fallback: per-lane cooperative gather into LDS (b128 loads, same layout).
  for (int k = 0; k < K_DIM; ++k) {
    const float* src = x + ((size_t)b * P_DIM + (size_t)rt[k]) * D_DIM;
    *(float4*)&tile[w][k][lane * 4]       = *(const float4*)&src[lane * 4];
    *(float4*)&tile[w][k][128 + lane * 4] = *(const float4*)&src[128 + lane * 4];
  }
#endif

  // ---- Weighted reduction: 32 ds_load_b128 + FMAs with SGPR weights -------
  const int c0 = lane * 4;
  const int c1 = 128 + lane * 4;
  float a0 = 0.f, a1 = 0.f, a2 = 0.f, a3 = 0.f;
  float a4 = 0.f, a5 = 0.f, a6 = 0.f, a7 = 0.f;

#pragma unroll
  for (int k = 0; k < K_DIM; ++k) {
    const float wk = wt[k];                  // wave-uniform -> SGPR operand
    const float4 lo = *(const float4*)&tile[w][k][c0];
    const float4 hi = *(const float4*)&tile[w][k][c1];
    a0 += wk * lo.x;  a1 += wk * lo.y;  a2 += wk * lo.z;  a3 += wk * lo.w;
    a4 += wk * hi.x;  a5 += wk * hi.y;  a6 += wk * hi.z;  a7 += wk * hi.w;
  }

  float* op = out + (size_t)g * D_DIM;
  float4 r0; r0.x = a0; r0.y = a1; r0.z = a2; r0.w = a3;
  float4 r1; r1.x = a4; r1.y = a5; r1.z = a6; r1.w = a7;
  *(float4*)&op[c0] = r0;                    // coalesced global_store_b128
  *(float4*)&op[c1] = r1;
}

extern "C" void kernel_launch(void* const* d_in, const int* in_sizes, int n_in,
                              void* d_out, int out_size, void* d_ws, size_t ws_size,
                              hipStream_t stream) {
  (void)in_sizes; (void)n_in; (void)d_ws; (void)ws_size; (void)out_size;
  const float* x       = (const float*)d_in[0];
  const int*   routes  = (const int*)d_in[1];
  const float* weights = (const float*)d_in[2];
  float*       out     = (float*)d_out;

  const int total_rows = B_DIM * P_DIM;                  // 32768 (b,p) pairs
  const int blocks     = total_rows / WAVES_PER_BLOCK;   // 8192 blocks
  wormhole_gather_tdm_kernel<<<blocks, 32 * WAVES_PER_BLOCK, 0, stream>>>(
      x, routes, weights, out);
}